// ScatterReduceAggregation_86311662780467
// MI455X (gfx1250) — compile-verified
//
#include <hip/hip_runtime.h>

typedef __attribute__((ext_vector_type(2))) float v2f;
typedef __attribute__((ext_vector_type(8))) float v8f;

#define D_COLS 128
#define ROWS_PER_BLOCK 512   // 128 groups of 4 rows per wave

// ---------------------------------------------------------------------------
// Phase 0: zero the output accumulator (harness poisons d_out with 0xAA).
// ---------------------------------------------------------------------------
__global__ void zero_out_kernel(float* __restrict__ out, int n) {
    int i = blockIdx.x * blockDim.x + threadIdx.x;
    int stride = gridDim.x * blockDim.x;
    for (; i < n; i += stride) out[i] = 0.0f;
}

// ---------------------------------------------------------------------------
// Phase 1: segmented column-sum via V_WMMA_F32_16X16X4_F32.
//
// Per block: 8 waves, wave w owns column slab [16w, 16w+16) and streams
// ROWS_PER_BLOCK rows in groups of 4 (the WMMA K dimension).
//   A (16x4)  : selector, A[slot,k] = (seg(row k) - seg_base == slot) ? 1 : 0
//   B (4x16)  : 4 rows x 16 cols of input data
//   C/D(16x16): per-slot partial column sums (8 VGPRs, 16 slots)
// VGPR layouts per ISA 7.12.2:
//   A: v0 = K=(lane<16?0:2), v1 = K+1, row M = lane&15
//   B: v0 = K=(lane<16?0:2), v1 = K+1, col N = lane&15   (rows striped by half)
//   D: vj = slot j (lanes 0-15) / slot j+8 (lanes 16-31), col N = lane&15
// Windows of 16 consecutive segment ids flush with f32 atomics; only nonzero
// slots touch memory. Sorted index => few flushes (~1 per 16 segments).
// TAIL=false: guard-free hot loop (no branch ladder, pure stream).
// ---------------------------------------------------------------------------
template <bool TAIL>
__device__ __forceinline__
void seg_sum_body(const float* __restrict__ inp,
                  const int*   __restrict__ idx,
                  float*       __restrict__ out,
                  int row0, int n_rows, int nseg,
                  int lane, int col) {
    const int m    = lane & 15;        // A row / D column within slab
    const int half = lane >> 4;        // 0: lanes 0-15, 1: lanes 16-31
    const int k0   = half * 2;         // K index held in vector element 0

    v8f acc = {};                      // 16 slots x 16 cols, all zero
    int seg_base = -1;                 // window not yet opened

    for (int g = 0; g < ROWS_PER_BLOCK; g += 4) {
        const int r = row0 + g;
        if (TAIL && r >= n_rows) break;

        // ---- uniform segment ids for the 4 rows (s_load_b128) ----
        int s0, s1, s2, s3;
        if (!TAIL || (r + 3 < n_rows)) {
            int4 s4 = *reinterpret_cast<const int4*>(idx + r);  // 16B aligned
            s0 = s4.x; s1 = s4.y; s2 = s4.z; s3 = s4.w;
        } else {
            s0 = idx[r];
            s1 = (r + 1 < n_rows) ? idx[r + 1] : -1;
            s2 = (r + 2 < n_rows) ? idx[r + 2] : -1;
            s3 = (r + 3 < n_rows) ? idx[r + 3] : -1;
        }
        const int slast = TAIL
            ? ((s3 >= 0) ? s3 : ((s2 >= 0) ? s2 : ((s1 >= 0) ? s1 : s0)))
            : s3;

        // ---- 16-slot window management (rarely-taken flush branch) ----
        if (seg_base < 0) seg_base = s0;
        if (slast - seg_base >= 16) {
            #pragma unroll
            for (int j = 0; j < 8; ++j) {
                float v = acc[j];
                int seg = seg_base + (half ? (j + 8) : j);
                if (v != 0.0f && seg < nseg)
                    atomicAdd(&out[(size_t)seg * D_COLS + col], v);
                acc[j] = 0.0f;
            }
            seg_base = s0;   // s3 - s0 <= 3 < 16, so the group fits now
        }

        // ---- load B: rows r+k0, r+k0+1 at this lane's column ----
        // (tail rows must read 0.0: WMMA turns 0 x garbage-NaN into NaN)
        const int rk0 = r + k0;
        const int rk1 = r + k0 + 1;
        v2f b;
        if (!TAIL) {
            b.x = inp[(size_t)rk0 * D_COLS + col];
            b.y = inp[(size_t)rk1 * D_COLS + col];
        } else {
            b.x = (rk0 < n_rows) ? inp[(size_t)rk0 * D_COLS + col] : 0.0f;
            b.y = (rk1 < n_rows) ? inp[(size_t)rk1 * D_COLS + col] : 0.0f;
        }

        // ---- build selector A (invalid rows: s=-1 never matches m) ----
        const int sk0 = (k0 == 0) ? s0 : s2;
        const int sk1 = (k0 == 0) ? s1 : s3;
        v2f a;
        a.x = (sk0 - seg_base == m) ? 1.0f : 0.0f;
        a.y = (sk1 - seg_base == m) ? 1.0f : 0.0f;

        // prefetch the stream ~32 rows ahead (global_prefetch_b8)
        if (!TAIL && (r + 32 < n_rows))
            __builtin_prefetch(&inp[(size_t)(r + 32) * D_COLS + col], 0, 1);

        // ---- D = A x B + C : exact f32 segmented accumulation ----
        acc = __builtin_amdgcn_wmma_f32_16x16x4_f32(
            /*neg_a=*/false, a, /*neg_b=*/false, b,
            /*c_mod=*/(short)0, acc, /*reuse_a=*/false, /*reuse_b=*/false);
    }

    // ---- final flush ----
    if (seg_base >= 0) {
        #pragma unroll
        for (int j = 0; j < 8; ++j) {
            float v = acc[j];
            int seg = seg_base + (half ? (j + 8) : j);
            if (v != 0.0f && seg < nseg)
                atomicAdd(&out[(size_t)seg * D_COLS + col], v);
        }
    }
}

__global__ __launch_bounds__(256)
void seg_sum_wmma_kernel(const float* __restrict__ inp,
                         const int*   __restrict__ idx,
                         float*       __restrict__ out,
                         int n_rows, int nseg) {
    const int wave = threadIdx.x >> 5;
    const int lane = threadIdx.x & 31;
    const int col  = wave * 16 + (lane & 15);   // global column this lane owns
    const int row0 = blockIdx.x * ROWS_PER_BLOCK;

    if (row0 + ROWS_PER_BLOCK <= n_rows) {
        // hot path: no guards anywhere in the stream loop
        seg_sum_body<false>(inp, idx, out, row0, n_rows, nseg, lane, col);
    } else {
        // final partial block only
        seg_sum_body<true>(inp, idx, out, row0, n_rows, nseg, lane, col);
    }
}

// ---------------------------------------------------------------------------
// Phase 2: divide by per-segment count. Index is sorted, so
// count[s] = lower_bound(s+1) - lower_bound(s); blockIdx.x == s is uniform,
// so the binary searches run on the scalar unit / constant cache.
// ---------------------------------------------------------------------------
__global__ __launch_bounds__(128)
void divide_by_count_kernel(float* __restrict__ out,
                            const int* __restrict__ idx,
                            int n_rows, int nseg) {
    const int s = blockIdx.x;
    if (s >= nseg) return;
    const int c = threadIdx.x;

    // first i with idx[i] >= s
    int lo = 0, hi = n_rows;
    while (lo < hi) { int mid = (lo + hi) >> 1; if (idx[mid] < s) lo = mid + 1; else hi = mid; }
    const int lb = lo;
    // first i with idx[i] >= s+1
    hi = n_rows;
    while (lo < hi) { int mid = (lo + hi) >> 1; if (idx[mid] < s + 1) lo = mid + 1; else hi = mid; }

    const float cnt = (float)(lo - lb);
    out[(size_t)s * D_COLS + c] /= fmaxf(cnt, 1.0f);
}

// ---------------------------------------------------------------------------
extern "C" void kernel_launch(void* const* d_in, const int* in_sizes, int n_in,
                              void* d_out, int out_size, void* d_ws, size_t ws_size,
                              hipStream_t stream) {
    const float* inp = (const float*)d_in[0];
    const int*   idx = (const int*)d_in[1];
    float*       out = (float*)d_out;

    const int n_rows = in_sizes[1];            // index length == #rows
    const int nseg   = out_size / D_COLS;      // 50,000

    // Phase 0: zero the accumulator
    {
        int n = out_size;
        int blocks = (n + 255) / 256;
        if (blocks > 16384) blocks = 16384;
        zero_out_kernel<<<blocks, 256, 0, stream>>>(out, n);
    }

    // Phase 1: WMMA segmented sum
    {
        int blocks = (n_rows + ROWS_PER_BLOCK - 1) / ROWS_PER_BLOCK;  // 3125
        seg_sum_wmma_kernel<<<blocks, 256, 0, stream>>>(inp, idx, out, n_rows, nseg);
    }

    // Phase 2: divide by counts (sorted-index binary search)
    {
        divide_by_count_kernel<<<nseg, D_COLS, 0, stream>>>(out, idx, n_rows, nseg);
    }
}